// NTMAttention_28750511080061
// MI455X (gfx1250) — compile-verified
//
#include <hip/hip_runtime.h>
#include <hip/hip_bf16.h>
#include <math.h>

typedef __attribute__((ext_vector_type(2))) float v2f;
typedef __attribute__((ext_vector_type(8))) float v8f;

#define B_DIM 1024
#define N_DIM 4096
#define M_DIM 64

__device__ __constant__ float EPS_COS_C = 1e-8f;

__device__ __forceinline__ float pick8(v8f v, int i) {
    float x = v[0];
    if (i == 1) x = v[1];
    if (i == 2) x = v[2];
    if (i == 3) x = v[3];
    if (i == 4) x = v[4];
    if (i == 5) x = v[5];
    if (i == 6) x = v[6];
    if (i == 7) x = v[7];
    return x;
}

// ---------------------------------------------------------------------------
// Phase 1: stream memory[B,N,M] once; per 16-row tile use f32 WMMA twice per
// 4-wide K chunk:  (A x kappa_bcast) -> dots,  (A x A) -> Gram diag = norms.
// Writes z[b,n] = beta[b] * cos(b,n) to workspace.
// ---------------------------------------------------------------------------
__global__ __launch_bounds__(256) void ntm_phase1_wmma(
    const float* __restrict__ memory,   // [B, N, M]
    const float* __restrict__ kappa,    // [B, M]
    const float* __restrict__ beta,     // [B]
    float* __restrict__ z)              // [B, N] workspace
{
    __shared__ float s_kappa[M_DIM];
    __shared__ float s_scale;           // beta[b] / max(||kappa||, eps)

    const int wgPerBatch = N_DIM / 256;                 // 16
    const int b     = blockIdx.x / wgPerBatch;
    const int nbase = (blockIdx.x % wgPerBatch) * 256;
    const int t = threadIdx.x;

    if (t < M_DIM) s_kappa[t] = kappa[b * M_DIM + t] + 1e-12f;
    __syncthreads();
    if (t == 0) {
        float ss = 0.f;
        #pragma unroll
        for (int i = 0; i < M_DIM; ++i) ss += s_kappa[i] * s_kappa[i];
        float kn = sqrtf(ss);
        if (kn < EPS_COS_C) kn = EPS_COS_C;
        s_scale = beta[b] / kn;
    }
    __syncthreads();

    const int lane = t & 31;
    const int wave = t >> 5;
    const int half = lane >> 4;      // 0: K={0,1}, 1: K={2,3} of each chunk
    const int r    = lane & 15;      // row within 16-row tile

    const float scale = s_scale;

    #pragma unroll
    for (int sub = 0; sub < 2; ++sub) {
        const int n0 = nbase + (wave * 2 + sub) * 16;
        const float* rowp =
            memory + ((size_t)b * N_DIM + (size_t)(n0 + r)) * M_DIM;

        v8f acc_d = {};   // dots   (broadcast columns)
        v8f acc_g = {};   // Gram   (diagonal = sum of squares)

        #pragma unroll
        for (int kb = 0; kb < M_DIM; kb += 4) {
            const int col = kb + half * 2;
            v2f a = *(const v2f*)(rowp + col);     // 8B load, 8B aligned
            a.x += 1e-12f;
            a.y += 1e-12f;
            v2f kf;
            kf.x = s_kappa[col];
            kf.y = s_kappa[col + 1];
            // D = A(16x4) x B(4x16) + C ; f32 full precision
            acc_d = __builtin_amdgcn_wmma_f32_16x16x4_f32(
                        false, a, false, kf, (short)0, acc_d, false, false);
            acc_g = __builtin_amdgcn_wmma_f32_16x16x4_f32(
                        false, a, false, a, (short)0, acc_g, false, false);
        }

        // Diagonal lanes {0..7, 24..31} hold D[j,j] in VGPR (lane&7).
        const bool active = (lane < 8) || (lane >= 24);
        const int idx = lane & 7;
        const float dv = pick8(acc_d, idx);
        const float gv = pick8(acc_g, idx);
        if (active) {
            float mn = sqrtf(gv);
            if (mn < EPS_COS_C) mn = EPS_COS_C;
            const int row = idx + ((lane >= 16) ? 8 : 0);
            z[(size_t)b * N_DIM + (size_t)(n0 + row)] = (dv / mn) * scale;
        }
    }
}

// ---------------------------------------------------------------------------
// Phase 2: per-batch softmax -> interpolate -> circular 3-tap shift ->
// sharpen -> normalize. One workgroup per batch, w_g lives in LDS.
// ---------------------------------------------------------------------------
__global__ __launch_bounds__(256) void ntm_phase2_finish(
    const float* __restrict__ z,        // [B, N]
    const float* __restrict__ g,        // [B]
    const float* __restrict__ s,        // [B, 3]
    const float* __restrict__ w_prev,   // [B, N]
    const float* __restrict__ gamma,    // [B]
    float* __restrict__ out)            // [B, N]
{
    __shared__ float s_wg[N_DIM];
    __shared__ float s_red[256];

    const int b = blockIdx.x;
    const int t = threadIdx.x;
    const int PER = N_DIM / 256;        // 16 elements / thread

    const float gv  = g[b];
    const float gam = gamma[b];
    const float s0  = s[b * 3 + 0];
    const float s1  = s[b * 3 + 1];
    const float s2  = s[b * 3 + 2];
    const float* zb = z + (size_t)b * N_DIM;
    const float* wp = w_prev + (size_t)b * N_DIM;

    // --- softmax: max ---
    float zi[16];
    float lmax = -INFINITY;
    #pragma unroll
    for (int i = 0; i < PER; ++i) {
        zi[i] = zb[t + i * 256];
        lmax = fmaxf(lmax, zi[i]);
    }
    s_red[t] = lmax;
    __syncthreads();
    for (int off = 128; off > 0; off >>= 1) {
        if (t < off) s_red[t] = fmaxf(s_red[t], s_red[t + off]);
        __syncthreads();
    }
    const float zmax = s_red[0];
    __syncthreads();

    // --- softmax: exp & sum ---
    float lsum = 0.f;
    #pragma unroll
    for (int i = 0; i < PER; ++i) {
        float e = __expf(zi[i] - zmax);
        s_wg[t + i * 256] = e;
        lsum += e;
    }
    s_red[t] = lsum;
    __syncthreads();
    for (int off = 128; off > 0; off >>= 1) {
        if (t < off) s_red[t] += s_red[t + off];
        __syncthreads();
    }
    const float inv_esum = 1.0f / s_red[0];
    __syncthreads();

    // --- interpolate into LDS ---
    #pragma unroll
    for (int i = 0; i < PER; ++i) {
        const int n = t + i * 256;
        const float wc = s_wg[n] * inv_esum;
        s_wg[n] = gv * wc + (1.0f - gv) * wp[n];
    }
    __syncthreads();

    // --- circular 3-tap shift + sharpen ---
    float wi[16];
    float lsum2 = 0.f;
    #pragma unroll
    for (int i = 0; i < PER; ++i) {
        const int n  = t + i * 256;
        const int nm = (n == 0) ? (N_DIM - 1) : (n - 1);
        const int np = (n == N_DIM - 1) ? 0 : (n + 1);
        const float wh = s0 * s_wg[nm] + s1 * s_wg[n] + s2 * s_wg[np];
        const float w  = powf(wh, gam);
        wi[i] = w;
        lsum2 += w;
    }
    s_red[t] = lsum2;
    __syncthreads();
    for (int off = 128; off > 0; off >>= 1) {
        if (t < off) s_red[t] += s_red[t + off];
        __syncthreads();
    }
    const float invw = 1.0f / (s_red[0] + 1e-12f);

    float* ob = out + (size_t)b * N_DIM;
    #pragma unroll
    for (int i = 0; i < PER; ++i) ob[t + i * 256] = wi[i] * invw;
}

extern "C" void kernel_launch(void* const* d_in, const int* in_sizes, int n_in,
                              void* d_out, int out_size, void* d_ws, size_t ws_size,
                              hipStream_t stream) {
    const float* beta   = (const float*)d_in[0];   // [B,1]
    const float* kappa  = (const float*)d_in[1];   // [B,M]
    const float* gamma  = (const float*)d_in[2];   // [B,1]
    const float* g      = (const float*)d_in[3];   // [B,1]
    const float* s      = (const float*)d_in[4];   // [B,3]
    const float* w_prev = (const float*)d_in[5];   // [B,N]
    const float* memory = (const float*)d_in[6];   // [B,N,M]
    float* out = (float*)d_out;
    float* z   = (float*)d_ws;                     // [B,N] = 16 MB scratch

    (void)in_sizes; (void)n_in; (void)out_size; (void)ws_size;

    const int grid1 = B_DIM * (N_DIM / 256);       // 16384 workgroups
    ntm_phase1_wmma<<<grid1, 256, 0, stream>>>(memory, kappa, beta, z);
    ntm_phase2_finish<<<B_DIM, 256, 0, stream>>>(z, g, s, w_prev, gamma, out);
}